// DGCNLayer_67327907332630
// MI455X (gfx1250) — compile-verified
//
#include <hip/hip_runtime.h>
#include <math.h>

#define NUU 100000
#define NII 200000
#define NE  1200000
#define DD  64

typedef float v2f __attribute__((ext_vector_type(2)));
typedef float v8f __attribute__((ext_vector_type(8)));

__device__ __forceinline__ float leakyf(float x) { return x >= 0.f ? x : 0.2f * x; }
// order-preserving float<->uint mapping for atomicMax on floats
__device__ __forceinline__ unsigned ordf(float x) {
  unsigned b = __float_as_uint(x);
  return (b & 0x80000000u) ? ~b : (b | 0x80000000u);
}
__device__ __forceinline__ float unordf(unsigned u) {
  unsigned b = (u & 0x80000000u) ? (u & 0x7FFFFFFFu) : ~u;
  return __uint_as_float(b);
}

// ---------------- fill kernels ----------------
__global__ void k_fill_f32(float* p, float v, int n) {
  int i = blockIdx.x * blockDim.x + threadIdx.x;
  if (i < n) p[i] = v;
}
__global__ void k_fill_u32(unsigned* p, unsigned v, int n) {
  int i = blockIdx.x * blockDim.x + threadIdx.x;
  if (i < n) p[i] = v;
}
__global__ void k_fill_bias_rows(float* p, const float* __restrict__ b, int n) {
  int i = blockIdx.x * blockDim.x + threadIdx.x;
  if (i < n) p[i] = b[i & 63];
}

// ---------------- WMMA fp32 GEMM: C[M,64] = A[M,64] @ W[64,64] ----------------
// wave32: each wave computes a 16x64 tile with V_WMMA_F32_16X16X4_F32.
// A frag (16x4 f32): lanes 0-15 hold row M=lane, K=k0..k0+1; lanes 16-31 hold K=k0+2..k0+3.
// B frag (4x16 f32): v0 = rows K=k0 (lanes 0-15) / K=k0+2 (lanes 16-31); v1 = K=k0+1 / K=k0+3.
// C/D (16x16 f32): VGPR j, lanes 0-15 -> M=j, lanes 16-31 -> M=j+8.
__global__ void k_gemm64(const float* __restrict__ A, const float* __restrict__ W,
                         float* __restrict__ C, int M) {
  __shared__ float Ws[64 * 64];
  for (int i = threadIdx.x; i < 64 * 64; i += blockDim.x) Ws[i] = W[i];
  __syncthreads();
  int wave = threadIdx.x >> 5, lane = threadIdx.x & 31;
  int tile = blockIdx.x * 8 + wave;              // 16-row tile index (wave-uniform)
  if (tile * 16 >= M) return;                    // uniform exit: EXEC stays all-1s for WMMA
  int mrow  = tile * 16 + (lane & 15);
  int khalf = (lane >> 4) * 2;                   // 0 for lanes 0-15, 2 for lanes 16-31
  v8f acc[4] = {};
  for (int kk = 0; kk < 64; kk += 4) {
    v2f a;
    a.x = A[(size_t)mrow * 64 + kk + khalf];
    a.y = A[(size_t)mrow * 64 + kk + khalf + 1];
#pragma unroll
    for (int t = 0; t < 4; ++t) {
      v2f b;
      int n = t * 16 + (lane & 15);
      b.x = Ws[(kk + khalf) * 64 + n];
      b.y = Ws[(kk + khalf + 1) * 64 + n];
      acc[t] = __builtin_amdgcn_wmma_f32_16x16x4_f32(false, a, false, b,
                                                     (short)0, acc[t], false, false);
    }
  }
  int rbase = tile * 16 + ((lane >> 4) << 3);
#pragma unroll
  for (int t = 0; t < 4; ++t) {
    int col = t * 16 + (lane & 15);
#pragma unroll
    for (int j = 0; j < 8; ++j)
      C[(size_t)(rbase + j) * 64 + col] = acc[t][j];
  }
}

// ---------------- WMMA fp32 GEMM: C = relu([A1|A2][M,128] @ W[128,64] + bias) ----------------
__global__ void k_gemm_concat_relu(const float* __restrict__ A1, const float* __restrict__ A2,
                                   const float* __restrict__ W, const float* __restrict__ bias,
                                   float* __restrict__ C, int M) {
  __shared__ float Ws[128 * 64];
  for (int i = threadIdx.x; i < 128 * 64; i += blockDim.x) Ws[i] = W[i];
  __syncthreads();
  int wave = threadIdx.x >> 5, lane = threadIdx.x & 31;
  int tile = blockIdx.x * 8 + wave;
  if (tile * 16 >= M) return;
  int mrow  = tile * 16 + (lane & 15);
  int khalf = (lane >> 4) * 2;
  v8f acc[4] = {};
  for (int kk = 0; kk < 128; kk += 4) {
    const float* Arow = (kk < 64) ? (A1 + (size_t)mrow * 64 + kk)
                                  : (A2 + (size_t)mrow * 64 + (kk - 64));
    v2f a;
    a.x = Arow[khalf];
    a.y = Arow[khalf + 1];
#pragma unroll
    for (int t = 0; t < 4; ++t) {
      v2f b;
      int n = t * 16 + (lane & 15);
      b.x = Ws[(kk + khalf) * 64 + n];
      b.y = Ws[(kk + khalf + 1) * 64 + n];
      acc[t] = __builtin_amdgcn_wmma_f32_16x16x4_f32(false, a, false, b,
                                                     (short)0, acc[t], false, false);
    }
  }
  int rbase = tile * 16 + ((lane >> 4) << 3);
#pragma unroll
  for (int t = 0; t < 4; ++t) {
    int col = t * 16 + (lane & 15);
    float bb = bias[col];
#pragma unroll
    for (int j = 0; j < 8; ++j) {
      float v = acc[t][j] + bb;
      C[(size_t)(rbase + j) * 64 + col] = v > 0.f ? v : 0.f;
    }
  }
}

// ---------------- s[row] = H[row,:] . a[0:64]  (one wave per row) ----------------
__global__ void k_rowdot(const float* __restrict__ H, const float* __restrict__ av,
                         float* __restrict__ s, int M) {
  int row  = blockIdx.x * (blockDim.x >> 5) + (threadIdx.x >> 5);
  int lane = threadIdx.x & 31;
  if (row >= M) return;
  float v = H[(size_t)row * 64 + lane] * av[lane] +
            H[(size_t)row * 64 + 32 + lane] * av[32 + lane];
  for (int o = 16; o > 0; o >>= 1) v += __shfl_xor(v, o, 32);
  if (lane == 0) s[row] = v;
}

// ---------------- GCN scatter: out[dst] += w[e] * sup[src]  (thread per edge-feature) ----------------
__global__ void k_scatter_gcn(const float* __restrict__ sup, const int* __restrict__ srcI,
                              const int* __restrict__ dstI, const float* __restrict__ w,
                              float* out, int total) {
  int i = blockIdx.x * blockDim.x + threadIdx.x;
  if (i >= total) return;
  int e = i >> 6, d = i & 63;
  atomicAdd(&out[(size_t)dstI[e] * 64 + d], w[e] * sup[(size_t)srcI[e] * 64 + d]);
}

// ---------------- GAT edge passes ----------------
__global__ void k_edge_max(const float* __restrict__ sd, const float* __restrict__ ss,
                           const int* __restrict__ dstI, const int* __restrict__ srcI,
                           unsigned* m, int ne) {
  int e = blockIdx.x * blockDim.x + threadIdx.x;
  if (e >= ne) return;
  float v = leakyf(sd[dstI[e]] + ss[srcI[e]]);
  atomicMax(&m[dstI[e]], ordf(v));
}
__global__ void k_edge_exp(const float* __restrict__ sd, const float* __restrict__ ss,
                           const int* __restrict__ dstI, const int* __restrict__ srcI,
                           const unsigned* __restrict__ m, float* ex, float* den, int ne) {
  int e = blockIdx.x * blockDim.x + threadIdx.x;
  if (e >= ne) return;
  int t = dstI[e];
  float v = leakyf(sd[t] + ss[srcI[e]]);
  float x = expf(v - unordf(m[t]));
  ex[e] = x;
  atomicAdd(&den[t], x);
}
__global__ void k_scatter_gat(const float* __restrict__ hsrc, const int* __restrict__ srcI,
                              const int* __restrict__ dstI, const float* __restrict__ ex,
                              const float* __restrict__ den, float* out, int total) {
  int i = blockIdx.x * blockDim.x + threadIdx.x;
  if (i >= total) return;
  int e = i >> 6, d = i & 63;
  int t = dstI[e];
  float alpha = ex[e] / (den[t] + 1e-16f);
  atomicAdd(&out[(size_t)t * 64 + d], alpha * hsrc[(size_t)srcI[e] * 64 + d]);
}

static inline int gb(int n, int t) { return (n + t - 1) / t; }

extern "C" void kernel_launch(void* const* d_in, const int* in_sizes, int n_in,
                              void* d_out, int out_size, void* d_ws, size_t ws_size,
                              hipStream_t stream) {
  (void)in_sizes; (void)n_in; (void)out_size; (void)ws_size;
  const float* ufea     = (const float*)d_in[0];
  const float* vfea     = (const float*)d_in[1];
  const int*   edge_user= (const int*)d_in[2];
  const int*   edge_item= (const int*)d_in[3];
  const float* uv_w     = (const float*)d_in[4];
  const float* vu_w     = (const float*)d_in[5];
  const float* W_gc1    = (const float*)d_in[6];
  const float* b_gc1    = (const float*)d_in[7];
  const float* W_gc2    = (const float*)d_in[8];
  const float* b_gc2    = (const float*)d_in[9];
  const float* W_gat3   = (const float*)d_in[10];
  const float* a_gat3   = (const float*)d_in[11];
  const float* W_gat4   = (const float*)d_in[12];
  const float* a_gat4   = (const float*)d_in[13];
  const float* W_user   = (const float*)d_in[14];
  const float* b_user   = (const float*)d_in[15];
  const float* W_item   = (const float*)d_in[16];
  const float* b_item   = (const float*)d_in[17];

  // workspace (floats), buffers reused across sequential phases
  float* ws = (float*)d_ws;
  size_t off = 0;
  float* P  = ws + off; off += (size_t)NUU * DD;  // supportU -> h_dst3 -> h_src4
  float* Q  = ws + off; off += (size_t)NII * DD;  // user_ho_i -> h_dst4
  float* R  = ws + off; off += (size_t)NII * DD;  // supportV -> h_src3 -> item_ho
  float* S  = ws + off; off += (size_t)NUU * DD;  // item_ho_u
  float* T  = ws + off; off += (size_t)NUU * DD;  // user_ho
  float* sd = ws + off; off += (size_t)NII;       // s_dst (max size)
  float* ss = ws + off; off += (size_t)NII;       // s_src (max size)
  unsigned* m = (unsigned*)(ws + off); off += (size_t)NII;
  float* den = ws + off; off += (size_t)NII;
  float* ex  = ws + off; off += (size_t)NE;

  float* outU = (float*)d_out;
  float* outV = outU + (size_t)NUU * DD;

  const int EF = NE * DD;  // 76.8M edge-feature work items

  // ---- GCN hop 1: user_ho_i = spmm(VU, ufea@W_gc1) + b_gc1   [NI,64] in Q
  k_gemm64<<<gb(NUU / 16, 8), 256, 0, stream>>>(ufea, W_gc1, P, NUU);
  k_fill_bias_rows<<<gb(NII * DD, 256), 256, 0, stream>>>(Q, b_gc1, NII * DD);
  k_scatter_gcn<<<gb(EF, 256), 256, 0, stream>>>(P, edge_user, edge_item, vu_w, Q, EF);

  // ---- GCN hop 2: item_ho_u = spmm(UV, vfea@W_gc2) + b_gc2   [NU,64] in S
  k_gemm64<<<gb(NII / 16, 8), 256, 0, stream>>>(vfea, W_gc2, R, NII);
  k_fill_bias_rows<<<gb(NUU * DD, 256), 256, 0, stream>>>(S, b_gc2, NUU * DD);
  k_scatter_gcn<<<gb(EF, 256), 256, 0, stream>>>(R, edge_item, edge_user, uv_w, S, EF);

  // ---- GAT3: dst=users(ufea), src=items(user_ho_i=Q); src idx=edge_item, dst idx=edge_user
  k_gemm64<<<gb(NUU / 16, 8), 256, 0, stream>>>(ufea, W_gat3, P, NUU);  // h_dst3
  k_gemm64<<<gb(NII / 16, 8), 256, 0, stream>>>(Q, W_gat3, R, NII);     // h_src3
  k_rowdot<<<gb(NUU, 8), 256, 0, stream>>>(P, a_gat3, sd, NUU);         // s_dst3
  k_rowdot<<<gb(NII, 8), 256, 0, stream>>>(R, a_gat3 + DD, ss, NII);    // s_src3
  k_fill_u32<<<gb(NUU, 256), 256, 0, stream>>>(m, 0u, NUU);
  k_edge_max<<<gb(NE, 256), 256, 0, stream>>>(sd, ss, edge_user, edge_item, m, NE);
  k_fill_f32<<<gb(NUU, 256), 256, 0, stream>>>(den, 0.f, NUU);
  k_edge_exp<<<gb(NE, 256), 256, 0, stream>>>(sd, ss, edge_user, edge_item, m, ex, den, NE);
  k_fill_f32<<<gb(NUU * DD, 256), 256, 0, stream>>>(T, 0.f, NUU * DD);
  k_scatter_gat<<<gb(EF, 256), 256, 0, stream>>>(R, edge_item, edge_user, ex, den, T, EF);

  // ---- GAT4: dst=items(vfea), src=users(item_ho_u=S); src idx=edge_user, dst idx=edge_item
  k_gemm64<<<gb(NII / 16, 8), 256, 0, stream>>>(vfea, W_gat4, Q, NII);  // h_dst4
  k_gemm64<<<gb(NUU / 16, 8), 256, 0, stream>>>(S, W_gat4, P, NUU);     // h_src4
  k_rowdot<<<gb(NII, 8), 256, 0, stream>>>(Q, a_gat4, sd, NII);         // s_dst4
  k_rowdot<<<gb(NUU, 8), 256, 0, stream>>>(P, a_gat4 + DD, ss, NUU);    // s_src4
  k_fill_u32<<<gb(NII, 256), 256, 0, stream>>>(m, 0u, NII);
  k_edge_max<<<gb(NE, 256), 256, 0, stream>>>(sd, ss, edge_item, edge_user, m, NE);
  k_fill_f32<<<gb(NII, 256), 256, 0, stream>>>(den, 0.f, NII);
  k_edge_exp<<<gb(NE, 256), 256, 0, stream>>>(sd, ss, edge_item, edge_user, m, ex, den, NE);
  k_fill_f32<<<gb(NII * DD, 256), 256, 0, stream>>>(R, 0.f, NII * DD);
  k_scatter_gat<<<gb(EF, 256), 256, 0, stream>>>(P, edge_user, edge_item, ex, den, R, EF);

  // ---- final: relu(concat @ W + b)
  k_gemm_concat_relu<<<gb(NUU / 16, 8), 256, 0, stream>>>(T, ufea, W_user, b_user, outU, NUU);
  k_gemm_concat_relu<<<gb(NII / 16, 8), 256, 0, stream>>>(R, vfea, W_item, b_item, outV, NII);
}